// GNN_72078141162181
// MI455X (gfx1250) — compile-verified
//
#include <hip/hip_runtime.h>

typedef float v2f __attribute__((ext_vector_type(2)));
typedef float v8f __attribute__((ext_vector_type(8)));

#define K_DIM 128

// ---------------- utility kernels ----------------

__global__ void gnn_zero_f32(float* __restrict__ p, long n) {
    long i = (long)blockIdx.x * blockDim.x + threadIdx.x;
    long stride = (long)gridDim.x * blockDim.x;
    for (; i < n; i += stride) p[i] = 0.0f;
}

__global__ void gnn_degree_count(const int* __restrict__ src, const int* __restrict__ dst,
                                 float* __restrict__ degO, float* __restrict__ degI, int E) {
    int e = blockIdx.x * blockDim.x + threadIdx.x;
    if (e < E) {
        atomicAdd(&degO[src[e]], 1.0f);
        atomicAdd(&degI[dst[e]], 1.0f);
    }
}

__global__ void gnn_rsqrt_clamp(float* __restrict__ p, int n) {
    int i = blockIdx.x * blockDim.x + threadIdx.x;
    if (i < n) p[i] = rsqrtf(fmaxf(p[i], 1.0f));
}

// agg[dst] += h[src], one wave per edge, 4 floats per lane (128 features)
__global__ void gnn_scatter_add(const int* __restrict__ src, const int* __restrict__ dst,
                                const float* __restrict__ hin, float* __restrict__ agg, int E) {
    int gid = blockIdx.x * blockDim.x + threadIdx.x;
    int e = gid >> 5;
    int lane = gid & 31;
    if (e >= E) return;
    int s = src[e];
    int d = dst[e];
    const float4 v = *(const float4*)(hin + (size_t)s * K_DIM + lane * 4);
    float* o = agg + (size_t)d * K_DIM + lane * 4;
    atomicAdd(o + 0, v.x);
    atomicAdd(o + 1, v.y);
    atomicAdd(o + 2, v.z);
    atomicAdd(o + 3, v.w);
}

// out = (optional relu)(agg * rin[row] + b[col]), 128 features per row
template <bool RELU>
__global__ void gnn_finalize(const float* __restrict__ agg, const float* __restrict__ rin,
                             const float* __restrict__ b, float* __restrict__ out, long n) {
    long i = (long)blockIdx.x * blockDim.x + threadIdx.x;
    if (i < n) {
        int m = (int)(i >> 7);          // row (feature dim == 128)
        float v = agg[i] * rin[m] + b[(int)(i & 127)];
        out[i] = RELU ? fmaxf(v, 0.0f) : v;
    }
}

// ---------------- WMMA fp32 GEMM ----------------
// out[M x N_OUT] = (A * rowscale) @ W  (+ bias), K = 128.
// Block: 16 rows x N_OUT cols; one wave per 16-col tile; W staged in LDS.
// A fragment layout (ISA 7.12.2, 32-bit A 16x4): lanes 0-15 -> K0/K1, lanes 16-31 -> K2/K3.
// C/D layout: VGPR i holds M = i (lanes 0-15) / M = 8+i (lanes 16-31), N = lane%16.
template <int N_OUT, bool HAS_SCALE, bool HAS_BIAS>
__global__ __launch_bounds__(2 * N_OUT) void gnn_gemm_wmma(
    const float* __restrict__ A, const float* __restrict__ scale,
    const float* __restrict__ W, const float* __restrict__ bias,
    float* __restrict__ out) {
    __shared__ float Wlds[K_DIM * N_OUT];

    // cooperative load of W (K_DIM x N_OUT, row-major k-major) into LDS
    {
        const float4* wsrc = (const float4*)W;
        float4* wdst = (float4*)Wlds;
        const int total4 = (K_DIM * N_OUT) / 4;
        for (int i = threadIdx.x; i < total4; i += blockDim.x) wdst[i] = wsrc[i];
    }
    __syncthreads();

    const int m0 = blockIdx.x * 16;
    const int wave = threadIdx.x >> 5;
    const int lane = threadIdx.x & 31;
    const int half = lane >> 4;     // 0: K0/K1 half, 1: K2/K3 half
    const int l = lane & 15;
    const int n0 = wave * 16;
    const int row = m0 + l;
    const int kb = half * 2;

    float srow = 1.0f;
    if (HAS_SCALE) srow = scale[row];

    const float* arow = A + (size_t)row * K_DIM + kb;

    v8f acc = {};
#pragma unroll
    for (int s = 0; s < K_DIM / 4; ++s) {
        const int kk = 4 * s + kb;
        float2 av = *(const float2*)(arow + 4 * s);   // A[row][kk], A[row][kk+1]
        v2f a;
        a.x = av.x * srow;
        a.y = av.y * srow;
        v2f b;
        b.x = Wlds[kk * N_OUT + n0 + l];              // W[kk][n0+l]
        b.y = Wlds[(kk + 1) * N_OUT + n0 + l];        // W[kk+1][n0+l]
        acc = __builtin_amdgcn_wmma_f32_16x16x4_f32(
            /*neg_a=*/false, a, /*neg_b=*/false, b,
            /*c_mod=*/(short)0, acc, /*reuse_a=*/false, /*reuse_b=*/false);
    }

    float bv = 0.0f;
    if (HAS_BIAS) bv = bias[n0 + l];
#pragma unroll
    for (int i = 0; i < 8; ++i) {
        out[(size_t)(m0 + half * 8 + i) * N_OUT + n0 + l] = acc[i] + bv;
    }
}

// ---------------- launcher ----------------

extern "C" void kernel_launch(void* const* d_in, const int* in_sizes, int n_in,
                              void* d_out, int out_size, void* d_ws, size_t ws_size,
                              hipStream_t stream) {
    const float* x  = (const float*)d_in[0];
    const int*   src = (const int*)d_in[1];
    const int*   dst = (const int*)d_in[2];
    const float* W1 = (const float*)d_in[3];
    const float* b1 = (const float*)d_in[4];
    const float* W2 = (const float*)d_in[5];
    const float* b2 = (const float*)d_in[6];
    const float* Wf = (const float*)d_in[7];
    const float* bf = (const float*)d_in[8];
    float* out = (float*)d_out;

    const int N = in_sizes[0] / K_DIM;   // 50000
    const int E = in_sizes[1];           // 1600000

    float* degO = (float*)d_ws;                 // N        (rsqrt(deg_out))
    float* degI = degO + N;                     // N        (rsqrt(deg_in))
    float* bufA = degI + N;                     // N * 128
    float* bufB = bufA + (size_t)N * K_DIM;     // N * 128

    const long nh = (long)N * K_DIM;
    const int mtiles = N / 16;                          // 3125 (N divisible by 16)
    const int scatterBlocks = (E * 32 + 255) / 256;     // one wave per edge
    const int finBlocks = (int)((nh + 255) / 256);

    // degrees: count, then in-place rsqrt(max(d,1)) over both arrays
    gnn_zero_f32<<<1024, 256, 0, stream>>>(degO, 2L * N);
    gnn_degree_count<<<(E + 255) / 256, 256, 0, stream>>>(src, dst, degO, degI, E);
    gnn_rsqrt_clamp<<<(2 * N + 255) / 256, 256, 0, stream>>>(degO, 2 * N);

    // ---- Layer 1: bufA = (x * degO) @ W1; bufB = scatter(bufA); bufA = relu(bufB*degI + b1)
    gnn_gemm_wmma<128, true, false><<<mtiles, 256, 0, stream>>>(x, degO, W1, nullptr, bufA);
    gnn_zero_f32<<<4096, 256, 0, stream>>>(bufB, nh);
    gnn_scatter_add<<<scatterBlocks, 256, 0, stream>>>(src, dst, bufA, bufB, E);
    gnn_finalize<true><<<finBlocks, 256, 0, stream>>>(bufB, degI, b1, bufA, nh);

    // ---- Layer 2: bufB = (bufA * degO) @ W2; bufA = scatter(bufB); bufB = bufA*degI + b2
    gnn_gemm_wmma<128, true, false><<<mtiles, 256, 0, stream>>>(bufA, degO, W2, nullptr, bufB);
    gnn_zero_f32<<<4096, 256, 0, stream>>>(bufA, nh);
    gnn_scatter_add<<<scatterBlocks, 256, 0, stream>>>(src, dst, bufB, bufA, E);
    gnn_finalize<false><<<finBlocks, 256, 0, stream>>>(bufA, degI, b2, bufB, nh);

    // ---- Final dense layer: out = bufB @ Wf + bf (N x 64)
    gnn_gemm_wmma<64, false, true><<<mtiles, 128, 0, stream>>>(bufB, nullptr, Wf, bf, out);
}